// HAKE_10179072491920
// MI455X (gfx1250) — compile-verified
//
#include <hip/hip_runtime.h>
#include <math.h>

#define NUM_ENTS 20000
#define DIM      256
#define BATCH    32
#define GAMMA    12.0f
// EMB_RANGE = (12+2)/256 ; phase radians = x*PI/(2*EMB_RANGE); v_sin_f32 computes
// sin(2*pi*input)  =>  fold 1/(2*pi): KPH = 1/(4*EMB_RANGE)
#define KPH (1.0f / (4.0f * (14.0f / 256.0f)))

typedef float v2f __attribute__((ext_vector_type(2)));
typedef float v8f __attribute__((ext_vector_type(8)));

// ---------------------------------------------------------------------------
// Prep: per-batch transformed rows into workspace (L2-resident, reused by all
// 313 main-kernel blocks).  hp = (ph_head+ph_rel)*KPH ; ac = -2*A*C ; c2 = C*C
// a2[b] = sum_d A^2
// ---------------------------------------------------------------------------
__global__ __launch_bounds__(256) void hake_prep(
    const int* __restrict__ e1, const int* __restrict__ rel,
    const float* __restrict__ emb_e, const float* __restrict__ emb_rel,
    float* __restrict__ hp, float* __restrict__ ac,
    float* __restrict__ c2, float* __restrict__ a2)
{
    const int b = blockIdx.x;
    const int d = threadIdx.x;
    const float* he = emb_e  + (size_t)e1[b]  * (2 * DIM);
    const float* re = emb_rel + (size_t)rel[b] * (3 * DIM);

    float phh = he[d];
    float mh  = he[DIM + d];
    float phr = re[d];
    float mr  = fabsf(re[DIM + d]);
    float br  = re[2 * DIM + d];
    br = fminf(br, 1.0f);
    br = (br < -mr) ? -mr : br;

    float A = mh * (mr + br);
    float C = 1.0f - br;

    hp[b * DIM + d] = (phh + phr) * KPH;
    ac[b * DIM + d] = -2.0f * A * C;
    c2[b * DIM + d] = C * C;

    __shared__ float red[256];
    red[d] = A * A;
    __syncthreads();
    #pragma unroll
    for (int s = 128; s > 0; s >>= 1) {
        if (d < s) red[d] += red[d + s];
        __syncthreads();
    }
    if (d == 0) a2[b] = red[0];
}

// ---------------------------------------------------------------------------
// Main: 8 waves/block.  wave -> 16x16 output tile (rows = batch, cols = ents).
//   brow0 = (wave&1)*16 ; ncol0 = blockIdx.x*64 + (wave>>1)*16
// r-term via two V_WMMA_F32_16X16X4_F32 per k-step; phase via v_sin_f32,
// accumulated directly in the WMMA C/D register layout:
//   VGPR v, lane L : M = v + 8*(L>=16), N = L&15
// ---------------------------------------------------------------------------
__global__ __launch_bounds__(256) void hake_main(
    const float* __restrict__ emb_e,
    const float* __restrict__ hp, const float* __restrict__ ac,
    const float* __restrict__ c2, const float* __restrict__ a2,
    const float* __restrict__ pw, const float* __restrict__ mw,
    float* __restrict__ out)
{
    const int tid  = threadIdx.x;
    const int wave = tid >> 5;
    const int lane = tid & 31;
    const int half = lane >> 4;   // 0: K lo pair / M 0..7 ; 1: K hi pair / M 8..15
    const int l16  = lane & 15;

    const int brow0 = (wave & 1) * 16;
    const int ncol0 = blockIdx.x * 64 + (wave >> 1) * 16;
    if (ncol0 >= NUM_ENTS) return;            // wave-uniform -> EXEC stays all-ones
    const int n = ncol0 + l16;

    const float* tailph  = emb_e + (size_t)n * (2 * DIM);        // phase part
    const float* tailmod = tailph + DIM;                          // modulus part
    const float* acrow = ac + (size_t)(brow0 + l16) * DIM + 2 * half;
    const float* c2row = c2 + (size_t)(brow0 + l16) * DIM + 2 * half;
    const float* hpbase = hp + (size_t)(brow0 + 8 * half) * DIM;

    v8f accr = {};                 // r-term GEMM accumulator (WMMA C layout)
    float ph[8];
    #pragma unroll
    for (int v = 0; v < 8; ++v) ph[v] = 0.0f;

    for (int k = 0; k < DIM; k += 4) {
        // prefetch next tail chunk (global_prefetch_b8)
        __builtin_prefetch(tailph + k + 64, 0, 0);

        // ---- r-term: D = (-2AC) * T  +  (C^2) * T^2  + ... --------------
        v2f aAC = *(const v2f*)(acrow + k);
        v2f aC2 = *(const v2f*)(c2row + k);
        v2f bT  = *(const v2f*)(tailmod + k + 2 * half);
        v2f bT2 = bT * bT;
        accr = __builtin_amdgcn_wmma_f32_16x16x4_f32(
                   false, aAC, false, bT,  (short)0, accr, false, false);
        accr = __builtin_amdgcn_wmma_f32_16x16x4_f32(
                   false, aC2, false, bT2, (short)0, accr, false, false);

        // ---- phase term: 32 sins per k-step (co-executes with WMMA) -----
        const float4 t4 = *(const float4*)(tailph + k);
        const float t0 = t4.x * KPH, t1 = t4.y * KPH,
                    t2 = t4.z * KPH, t3 = t4.w * KPH;
        #pragma unroll
        for (int v = 0; v < 8; ++v) {
            const float4 h4 = *(const float4*)(hpbase + v * DIM + k);
            ph[v] += fabsf(__builtin_amdgcn_sinf(h4.x - t0));
            ph[v] += fabsf(__builtin_amdgcn_sinf(h4.y - t1));
            ph[v] += fabsf(__builtin_amdgcn_sinf(h4.z - t2));
            ph[v] += fabsf(__builtin_amdgcn_sinf(h4.w - t3));
        }
    }

    const float pws = pw[0];
    const float mws = mw[0];
    #pragma unroll
    for (int v = 0; v < 8; ++v) {
        const int m = brow0 + v + 8 * half;                 // batch row
        float r2 = a2[m] + accr[v];                         // + sum(A^2)
        float rterm = sqrtf(fmaxf(r2, 0.0f)) * mws;
        float pterm = ph[v] * pws;
        float z = GAMMA - (pterm + rterm);
        out[(size_t)m * NUM_ENTS + n] = 1.0f / (1.0f + __expf(-z));
    }
}

// ---------------------------------------------------------------------------
extern "C" void kernel_launch(void* const* d_in, const int* in_sizes, int n_in,
                              void* d_out, int out_size, void* d_ws, size_t ws_size,
                              hipStream_t stream) {
    // inputs: 0:g 1:e1 2:rel 3:e2_multi(unused) 4:emb_e 5:emb_rel 6:pw 7:mw
    const int*   e1      = (const int*)d_in[1];
    const int*   rel     = (const int*)d_in[2];
    const float* emb_e   = (const float*)d_in[4];
    const float* emb_rel = (const float*)d_in[5];
    const float* pw      = (const float*)d_in[6];
    const float* mw      = (const float*)d_in[7];

    float* ws = (float*)d_ws;
    float* hp = ws;
    float* ac = ws + BATCH * DIM;
    float* c2 = ws + 2 * BATCH * DIM;
    float* a2 = ws + 3 * BATCH * DIM;   // 32 floats; total ws use ~96.1 KB

    hipLaunchKernelGGL(hake_prep, dim3(BATCH), dim3(256), 0, stream,
                       e1, rel, emb_e, emb_rel, hp, ac, c2, a2);

    const int nblocks = (NUM_ENTS + 63) / 64;   // 313
    hipLaunchKernelGGL(hake_main, dim3(nblocks), dim3(256), 0, stream,
                       emb_e, hp, ac, c2, a2, pw, mw, (float*)d_out);
}